// EntropicGCN_34041910788781
// MI455X (gfx1250) — compile-verified
//
#include <hip/hip_runtime.h>

typedef __attribute__((ext_vector_type(2))) float v2f;
typedef __attribute__((ext_vector_type(8))) float v8f;

#define DIN 128
#define DH  64
#define TEMPR 1.0f
#define WEIGHT 1.0f

struct Scalars {
    unsigned int mbits;   // bit pattern of max(e_raw) (e_raw >= 0 so uint-max == float-max)
    unsigned int pad;
    double Z, U, V;       // sum exp(z), sum exp(z)*eN, sum exp(z)*eN^2
};

// ---------------------------------------------------------------- utilities
__global__ void zero_f32_k(float* __restrict__ p, long long n) {
    long long i = (long long)blockIdx.x * blockDim.x + threadIdx.x;
    if (i < n) p[i] = 0.0f;
}

__global__ void fill_one_k(float* __restrict__ p, int n) {
    long long i = (long long)blockIdx.x * blockDim.x + threadIdx.x;
    if (i < n) p[i] = 1.0f;   // self-loop contribution to degree
}

__global__ void deg_edge_k(const int* __restrict__ dst, float* __restrict__ deg, long long e) {
    long long i = (long long)blockIdx.x * blockDim.x + threadIdx.x;
    if (i < e) atomicAdd(&deg[dst[i]], 1.0f);
}

__global__ void rsqrt_k(float* __restrict__ d, int n) {
    long long i = (long long)blockIdx.x * blockDim.x + threadIdx.x;
    if (i < n) d[i] = rsqrtf(d[i]);   // deg >= 1 always (self-loop)
}

// ---------------------------------------------------------------- WMMA GEMM
// C[n x 64] = A[n x K] * W[K x 64], fp32 via V_WMMA_F32_16X16X4_F32.
// 8 waves/block, 16 rows/wave -> 128 rows/block. W staged in LDS.
template <int K>
__global__ __launch_bounds__(256) void gemm_wmma_k(const float* __restrict__ A,
                                                   const float* __restrict__ W,
                                                   float* __restrict__ C, int n) {
    __shared__ float ldsW[K * DH];
    for (int i = threadIdx.x; i < K * DH; i += 256) ldsW[i] = W[i];
    __syncthreads();

    const int wave = threadIdx.x >> 5;
    const int lane = threadIdx.x & 31;
    const int rowBase = blockIdx.x * 128 + wave * 16;
    const int lrow = lane & 15;
    const int kgrp = (lane >> 4) << 1;          // lanes 0-15 -> K+0/K+1, lanes 16-31 -> K+2/K+3

    int arow = rowBase + lrow;
    if (arow > n - 1) arow = n - 1;             // clamp loads; stores are guarded
    const float* __restrict__ Arow = A + (long long)arow * K;

    v8f acc0 = {}, acc1 = {}, acc2 = {}, acc3 = {};
    for (int k = 0; k < K; k += 4) {
        v2f a;
        a.x = Arow[k + kgrp + 0];
        a.y = Arow[k + kgrp + 1];
        const float* w0 = &ldsW[(k + kgrp + 0) * DH + lrow];
        const float* w1 = &ldsW[(k + kgrp + 1) * DH + lrow];
        v2f b0; b0.x = w0[0];  b0.y = w1[0];
        v2f b1; b1.x = w0[16]; b1.y = w1[16];
        v2f b2; b2.x = w0[32]; b2.y = w1[32];
        v2f b3; b3.x = w0[48]; b3.y = w1[48];
        acc0 = __builtin_amdgcn_wmma_f32_16x16x4_f32(false, a, false, b0, (short)0, acc0, false, false);
        acc1 = __builtin_amdgcn_wmma_f32_16x16x4_f32(false, a, false, b1, (short)0, acc1, false, false);
        acc2 = __builtin_amdgcn_wmma_f32_16x16x4_f32(false, a, false, b2, (short)0, acc2, false, false);
        acc3 = __builtin_amdgcn_wmma_f32_16x16x4_f32(false, a, false, b3, (short)0, acc3, false, false);
    }

    const int rhalf = (lane >> 4) * 8;          // C/D layout: lane>=16 holds rows M=8..15
    for (int v = 0; v < 8; ++v) {
        int r = rowBase + rhalf + v;
        if (r < n) {
            float* Cr = C + (long long)r * DH + lrow;
            Cr[0]  = acc0[v];
            Cr[16] = acc1[v];
            Cr[32] = acc2[v];
            Cr[48] = acc3[v];
        }
    }
}

// ------------------------------------------------------- GCN aggregation
// out[i] = bias + dinv[i]^2 * hl[i]   (self-loop term, initializes out)
__global__ void agg_init_k(const float* __restrict__ hl, const float* __restrict__ bias,
                           const float* __restrict__ dinv, float* __restrict__ out, int n) {
    long long idx = (long long)blockIdx.x * blockDim.x + threadIdx.x;
    if (idx >= (long long)n * 16) return;
    int i = (int)(idx >> 4);
    int c = (int)(idx & 15);
    float w = dinv[i]; w *= w;
    float4 h = *((const float4*)(hl + (long long)i * DH) + c);
    float4 b = *((const float4*)bias + c);
    float4 o;
    o.x = b.x + w * h.x; o.y = b.y + w * h.y; o.z = b.z + w * h.z; o.w = b.w + w * h.w;
    *((float4*)(out + (long long)i * DH) + c) = o;
}

// out[dst] += dinv[src]*dinv[dst] * hl[src]  (one thread per edge x float4 chunk)
__global__ void agg_edge_k(const float* __restrict__ hl, const int* __restrict__ src,
                           const int* __restrict__ dst, const float* __restrict__ dinv,
                           float* __restrict__ out, long long total) {
    long long idx = (long long)blockIdx.x * blockDim.x + threadIdx.x;
    if (idx >= total) return;
    long long e = idx >> 4;
    int c = (int)(idx & 15);
    int s = src[e], d = dst[e];
    float coef = dinv[s] * dinv[d];
    float4 h = *((const float4*)(hl + (long long)s * DH) + c);
    float* op = out + (long long)d * DH + c * 4;
    atomicAdd(op + 0, coef * h.x);
    atomicAdd(op + 1, coef * h.y);
    atomicAdd(op + 2, coef * h.z);
    atomicAdd(op + 3, coef * h.w);
}

// ------------------------------------------------------- entropy gradient
// e_raw[src] += 0.5*||h[src]-h[dst]||^2   (16 lanes per edge, shfl reduce)
__global__ void energy_edge_k(const float* __restrict__ h, const int* __restrict__ src,
                              const int* __restrict__ dst, float* __restrict__ eraw, long long E) {
    long long idx = (long long)blockIdx.x * blockDim.x + threadIdx.x;
    long long e = idx >> 4;
    if (e >= E) return;
    int g = (int)(idx & 15);
    int s = src[e], d = dst[e];
    float4 a = *((const float4*)(h + (long long)s * DH) + g);
    float4 b = *((const float4*)(h + (long long)d * DH) + g);
    float dx = a.x - b.x, dy = a.y - b.y, dz = a.z - b.z, dw = a.w - b.w;
    float p = dx * dx + dy * dy + dz * dz + dw * dw;
    p += __shfl_xor(p, 8, 16);
    p += __shfl_xor(p, 4, 16);
    p += __shfl_xor(p, 2, 16);
    p += __shfl_xor(p, 1, 16);
    if (g == 0) atomicAdd(&eraw[s], 0.5f * p);
}

__global__ void reduce_max_k(const float* __restrict__ eraw, Scalars* __restrict__ sc, int n) {
    __shared__ float sm[256];
    int tid = threadIdx.x;
    long long i = (long long)blockIdx.x * 256 + tid;
    sm[tid] = (i < n) ? eraw[i] : 0.0f;
    __syncthreads();
    for (int s = 128; s > 0; s >>= 1) {
        if (tid < s) sm[tid] = fmaxf(sm[tid], sm[tid + s]);
        __syncthreads();
    }
    if (tid == 0) atomicMax(&sc->mbits, __float_as_uint(sm[0]));
}

__global__ void reduce_sums_k(const float* __restrict__ eraw, Scalars* __restrict__ sc, int n) {
    __shared__ float sZ[256], sU[256], sV[256];
    int tid = threadIdx.x;
    long long i = (long long)blockIdx.x * 256 + tid;
    float Mp = __uint_as_float(sc->mbits) + 1e-9f;
    float z = 0.f, u = 0.f, vv = 0.f;
    if (i < n) {
        float eN = eraw[i] / Mp;
        float ez = expf(-eN / TEMPR);      // z in [-1,0], no overflow concerns
        z = ez; u = ez * eN; vv = ez * eN * eN;
    }
    sZ[tid] = z; sU[tid] = u; sV[tid] = vv;
    __syncthreads();
    for (int s = 128; s > 0; s >>= 1) {
        if (tid < s) { sZ[tid] += sZ[tid + s]; sU[tid] += sU[tid + s]; sV[tid] += sV[tid + s]; }
        __syncthreads();
    }
    if (tid == 0) {
        atomicAdd(&sc->Z, (double)sZ[0]);
        atomicAdd(&sc->U, (double)sU[0]);
        atomicAdd(&sc->V, (double)sV[0]);
    }
}

// c_i = dS/d e_raw_i  (analytic grad of S through softmax + max-normalization)
__global__ void compute_c_k(const float* __restrict__ eraw, const Scalars* __restrict__ sc,
                            float* __restrict__ c, int n) {
    long long i = (long long)blockIdx.x * blockDim.x + threadIdx.x;
    if (i >= n) return;
    float Mf = __uint_as_float(sc->mbits);
    double Mp = (double)Mf + 1e-9;
    double Z = sc->Z, U = sc->U, V = sc->V;
    double A1 = U / Z;
    double logZ = log(Z);
    double S = A1 / (double)TEMPR + logZ;
    float er = eraw[i];
    double eN = (double)er / Mp;
    double p = exp(-eN / (double)TEMPR) / Z;
    double logp = -eN / (double)TEMPR - logZ;
    double q = p * (logp + S) / (double)TEMPR;
    double ci = q / Mp;
    if (__float_as_uint(er) == sc->mbits) {   // argmax correction from d(max)/d e_raw
        double A2 = V / Z;
        ci -= (A1 * A1 - A2) / ((double)TEMPR * (double)TEMPR * Mp);
    }
    c[i] = (float)ci;
}

// grad[src] += c[src]*(h_src - h_dst);  grad[dst] -= c[src]*(h_src - h_dst)
__global__ void grad_edge_k(const float* __restrict__ h, const int* __restrict__ src,
                            const int* __restrict__ dst, const float* __restrict__ c,
                            float* __restrict__ g, long long total) {
    long long idx = (long long)blockIdx.x * blockDim.x + threadIdx.x;
    if (idx >= total) return;
    long long e = idx >> 4;
    int ch = (int)(idx & 15);
    int s = src[e], d = dst[e];
    float w = c[s];
    float4 a = *((const float4*)(h + (long long)s * DH) + ch);
    float4 b = *((const float4*)(h + (long long)d * DH) + ch);
    float gx = w * (a.x - b.x), gy = w * (a.y - b.y), gz = w * (a.z - b.z), gw = w * (a.w - b.w);
    float* gs = g + (long long)s * DH + ch * 4;
    float* gd = g + (long long)d * DH + ch * 4;
    atomicAdd(gs + 0, gx);  atomicAdd(gs + 1, gy);  atomicAdd(gs + 2, gz);  atomicAdd(gs + 3, gw);
    atomicAdd(gd + 0, -gx); atomicAdd(gd + 1, -gy); atomicAdd(gd + 2, -gz); atomicAdd(gd + 3, -gw);
}

__global__ void relu_add_k(float* __restrict__ h, const float* __restrict__ g, long long n) {
    long long i = (long long)blockIdx.x * blockDim.x + threadIdx.x;
    if (i < n) h[i] = fmaxf(h[i] + WEIGHT * g[i], 0.0f);
}

// ---------------------------------------------------------------- launcher
extern "C" void kernel_launch(void* const* d_in, const int* in_sizes, int n_in,
                              void* d_out, int out_size, void* d_ws, size_t ws_size,
                              hipStream_t stream) {
    const float* x  = (const float*)d_in[0];
    const int*   ei = (const int*)d_in[1];
    const float* W1 = (const float*)d_in[2]; const float* b1 = (const float*)d_in[3];
    const float* W2 = (const float*)d_in[4]; const float* b2 = (const float*)d_in[5];
    const float* W3 = (const float*)d_in[6]; const float* b3 = (const float*)d_in[7];
    const float* Wo = (const float*)d_in[8]; const float* bo = (const float*)d_in[9];

    const int       N = in_sizes[0] / DIN;
    const long long E = in_sizes[1] / 2;
    const int* src = ei;
    const int* dst = ei + E;

    // workspace carve-out (256B aligned)
    char* ws = (char*)d_ws;
    size_t off = 0;
    auto carve = [&](size_t bytes) { size_t c = off; off += (bytes + 255) & ~(size_t)255; return c; };
    float*   lin  = (float*)(ws + carve((size_t)N * DH * 4));
    float*   hbuf = (float*)(ws + carve((size_t)N * DH * 4));
    float*   gbuf = (float*)(ws + carve((size_t)N * DH * 4));
    float*   dinv = (float*)(ws + carve((size_t)N * 4));
    float*   eraw = (float*)(ws + carve((size_t)N * 4));
    float*   cbuf = (float*)(ws + carve((size_t)N * 4));
    Scalars* sc   = (Scalars*)(ws + carve(256));
    (void)ws_size; (void)n_in; (void)out_size;

    const unsigned gN    = (unsigned)((N + 255) / 256);
    const unsigned gE    = (unsigned)((E + 255) / 256);
    const unsigned gN16  = (unsigned)(((long long)N * 16 + 255) / 256);
    const unsigned gE16  = (unsigned)((E * 16 + 255) / 256);
    const unsigned gNDH  = (unsigned)(((long long)N * DH + 255) / 256);
    const long long E16  = E * 16;

    // symmetric-normalization degrees (shared by all 4 convs)
    fill_one_k<<<gN, 256, 0, stream>>>(dinv, N);
    deg_edge_k<<<gE, 256, 0, stream>>>(dst, dinv, E);
    rsqrt_k<<<gN, 256, 0, stream>>>(dinv, N);

    auto conv = [&](const float* in, int K, const float* W, const float* bias, float* out) {
        if (K == DIN)
            gemm_wmma_k<DIN><<<(unsigned)((N + 127) / 128), 256, 0, stream>>>(in, W, lin, N);
        else
            gemm_wmma_k<DH><<<(unsigned)((N + 127) / 128), 256, 0, stream>>>(in, W, lin, N);
        agg_init_k<<<gN16, 256, 0, stream>>>(lin, bias, dinv, out, N);
        agg_edge_k<<<gE16, 256, 0, stream>>>(lin, src, dst, dinv, out, E16);
    };

    auto entropy_step = [&](float* h) {
        zero_f32_k<<<gN, 256, 0, stream>>>(eraw, N);
        zero_f32_k<<<1, 256, 0, stream>>>((float*)sc, 16);      // mbits=0, Z=U=V=0.0
        zero_f32_k<<<gNDH, 256, 0, stream>>>(gbuf, (long long)N * DH);
        energy_edge_k<<<gE16, 256, 0, stream>>>(h, src, dst, eraw, E);
        reduce_max_k<<<gN, 256, 0, stream>>>(eraw, sc, N);
        reduce_sums_k<<<gN, 256, 0, stream>>>(eraw, sc, N);
        compute_c_k<<<gN, 256, 0, stream>>>(eraw, sc, cbuf, N);
        grad_edge_k<<<gE16, 256, 0, stream>>>(h, src, dst, cbuf, gbuf, E16);
        relu_add_k<<<gNDH, 256, 0, stream>>>(h, gbuf, (long long)N * DH);
    };

    conv(x,    DIN, W1, b1, hbuf);  entropy_step(hbuf);
    conv(hbuf, DH,  W2, b2, hbuf);  entropy_step(hbuf);
    conv(hbuf, DH,  W3, b3, hbuf);  entropy_step(hbuf);
    conv(hbuf, DH,  Wo, bo, (float*)d_out);
}